// CausalSelfAttentionLTP_44538810860193
// MI455X (gfx1250) — compile-verified
//
#include <hip/hip_runtime.h>
#include <hip/hip_bf16.h>

// ---------------------------------------------------------------------------
// CausalSelfAttention (LTP) forward for MI455X / gfx1250.
// - All matrix math via V_WMMA_F32_16X16X4_F32 (native fp32 tensor op, exact).
// - Global->LDS staging via GLOBAL_LOAD_ASYNC_TO_LDS_B128 (ASYNCcnt tracked),
//   double-buffered, one barrier per pipeline iteration.
// - 16-query x 2048-key score slab lives entirely in LDS (exact softmax).
// ---------------------------------------------------------------------------

#define B_  4
#define T_  2048
#define C_  768
#define H_  12
#define HS_ 64

typedef float v2f __attribute__((ext_vector_type(2)));
typedef float v8f __attribute__((ext_vector_type(8)));

// ---- CDNA5 async global->LDS copy (16B per lane), ASYNCcnt tracked --------
__device__ __forceinline__ void async_b128(void* lds, const void* gaddr) {
  uint32_t l = (uint32_t)(uintptr_t)lds;     // low 32 bits = LDS byte address
  asm volatile("global_load_async_to_lds_b128 %0, %1, off"
               :: "v"(l), "v"(gaddr) : "memory");
}
__device__ __forceinline__ void wait_async0() {
  asm volatile("s_wait_asynccnt 0" ::: "memory");
}

// ---------------------------------------------------------------------------
// fp32 WMMA GEMM: C(MxN) = A(MxK) * B(KxN), row-major, M%64==N%128==K%32==0.
// 256 threads = 8 waves in a 2x4 grid; each wave owns a 32x32 patch (2x2
// WMMA tiles) of the 64x128 block tile -> 1.0 LDS ops per WMMA.
// ---------------------------------------------------------------------------
#define GBM 64
#define GBN 128
#define GKC 32

__global__ __launch_bounds__(256) void gemm_f32_wmma(
    const float* __restrict__ A, const float* __restrict__ Bm,
    float* __restrict__ Cm, int M, int N, int K) {
  __shared__ __align__(16) float As[2][GBM][GKC + 4];  // stride 36 (9x16B)
  __shared__ __align__(16) float Bs[2][GKC][GBN + 4];  // stride 132 (33x16B)

  const int tid  = threadIdx.x;
  const int lane = tid & 31;
  const int wave = tid >> 5;
  const int l15  = lane & 15;
  const int lhi  = lane >> 4;           // half-wave -> K pair {0,1} vs {2,3}

  const int m0 = blockIdx.y * GBM;
  const int n0 = blockIdx.x * GBN;
  const int wm = (wave & 1) * 32;       // wave's 32-row patch
  const int wn = (wave >> 1) * 32;      // wave's 32-col patch

  v8f acc00 = {}, acc01 = {}, acc10 = {}, acc11 = {};
  const int nk = K / GKC;

  auto issue_slab = [&](int k0, int buf) {
#pragma unroll
    for (int j = 0; j < 2; ++j) {       // A: 64x32 floats = 512 x 16B chunks
      int id = tid + j * 256;
      int r = id >> 3, c4 = id & 7;
      async_b128(&As[buf][r][c4 * 4], &A[(size_t)(m0 + r) * K + k0 + c4 * 4]);
    }
#pragma unroll
    for (int j = 0; j < 4; ++j) {       // B: 32x128 floats = 1024 x 16B chunks
      int id = tid + j * 256;
      int r = id >> 5, c4 = id & 31;
      async_b128(&Bs[buf][r][c4 * 4], &Bm[(size_t)(k0 + r) * N + n0 + c4 * 4]);
    }
  };

  issue_slab(0, 0);
  wait_async0();
  __syncthreads();
  for (int kc = 0; kc < nk; ++kc) {
    if (kc + 1 < nk) issue_slab((kc + 1) * GKC, (kc + 1) & 1);  // overlap DMA

    float (*Asb)[GKC + 4] = As[kc & 1];
    float (*Bsb)[GBN + 4] = Bs[kc & 1];
#pragma unroll
    for (int kk = 0; kk < GKC; kk += 4) {
      const int kq = kk + 2 * lhi;
      v2f a0 = *(const v2f*)&Asb[wm + l15][kq];        // contiguous b64
      v2f a1 = *(const v2f*)&Asb[wm + 16 + l15][kq];
      v2f b0, b1;
      b0.x = Bsb[kq][wn + l15];       b0.y = Bsb[kq + 1][wn + l15];
      b1.x = Bsb[kq][wn + 16 + l15];  b1.y = Bsb[kq + 1][wn + 16 + l15];
      acc00 = __builtin_amdgcn_wmma_f32_16x16x4_f32(false, a0, false, b0,
                                                    (short)0, acc00, false, false);
      acc01 = __builtin_amdgcn_wmma_f32_16x16x4_f32(false, a0, false, b1,
                                                    (short)0, acc01, false, false);
      acc10 = __builtin_amdgcn_wmma_f32_16x16x4_f32(false, a1, false, b0,
                                                    (short)0, acc10, false, false);
      acc11 = __builtin_amdgcn_wmma_f32_16x16x4_f32(false, a1, false, b1,
                                                    (short)0, acc11, false, false);
    }
    if (kc + 1 < nk) { wait_async0(); __syncthreads(); }
  }

#pragma unroll
  for (int r = 0; r < 8; ++r) {         // D: VGPR r, lane n -> D[r+8*lhi][n]
    int row0 = m0 + wm + r + 8 * lhi;
    int row1 = row0 + 16;
    Cm[(size_t)row0 * N + n0 + wn + l15]      = acc00[r];
    Cm[(size_t)row0 * N + n0 + wn + 16 + l15] = acc01[r];
    Cm[(size_t)row1 * N + n0 + wn + l15]      = acc10[r];
    Cm[(size_t)row1 * N + n0 + wn + 16 + l15] = acc11[r];
  }
}

// ---------------------------------------------------------------------------
__global__ void zero_f32(float* __restrict__ p, int n) {
  int i = blockIdx.x * blockDim.x + threadIdx.x;
  if (i < n) p[i] = 0.0f;
}

// ---------------------------------------------------------------------------
// Attention: one WG per (b, h, 16-query tile). Exact softmax over a
// 16 x 2048 LDS score slab. K/V slabs (128 keys x 64 dims) double-buffered
// and filled with async B128 copies. Row strides pad by 4 floats so WMMA
// fragment gathers (stride mod 64 == 4) hit 16 distinct LDS banks while
// keeping 16B alignment for the async copies. Q fragments are hoisted into
// registers once (loop-invariant across key slabs).
// ---------------------------------------------------------------------------
#define SSTR 2052                       // score row stride (2048 + 4)
#define KSTR 68                         // K/V/Q slab row stride (64 + 4)

__global__ __launch_bounds__(256) void attn_kernel(
    const float* __restrict__ qkv,      // (B*T, 3C)
    const int*   __restrict__ amask,    // (B, T)
    float* __restrict__ ws_y,           // (B*T, C) pre-projection output
    float* __restrict__ imp) {          // (B, T) importance accumulator
  extern __shared__ __align__(16) float smem[];
  float* Ssc  = smem;                        // 16  * SSTR
  float* Kbuf = Ssc + 16 * SSTR;             // 2 * 128 * KSTR (K, then V slabs)
  float* Qst  = Kbuf + 2 * 128 * KSTR;       // 16  * KSTR
  float* Yred = Qst + 16 * KSTR;             // 8 waves * 256

  const int tid  = threadIdx.x;
  const int lane = tid & 31;
  const int wave = tid >> 5;
  const int l15  = lane & 15;
  const int lhi  = lane >> 4;

  const int qt = blockIdx.x, h = blockIdx.y, b = blockIdx.z;
  const int qbase = qt * 16;
  const size_t bt   = (size_t)b * T_;
  const size_t rstr = 3 * C_;

  // async-issue one 128-key slab (K: part=1, V: part=2) into buffer `buf`
  auto issue_kv = [&](int keybase, int part, int buf) {
    float* dst = Kbuf + (size_t)buf * 128 * KSTR;
#pragma unroll
    for (int j = 0; j < 8; ++j) {       // 128 x 64 floats = 2048 x 16B chunks
      int id = tid + j * 256;
      int kk = id >> 4, c4 = id & 15;
      async_b128(&dst[kk * KSTR + c4 * 4],
                 &qkv[(bt + keybase + kk) * rstr + (size_t)part * C_ + h * HS_ + c4 * 4]);
    }
  };

  // ---- stage Q (16 x 64) + first K slab ----
  {
    int r = tid >> 4, c4 = tid & 15;    // 256 x 16B chunks
    async_b128(&Qst[r * KSTR + c4 * 4],
               &qkv[(bt + qbase + r) * rstr + h * HS_ + c4 * 4]);
  }
  issue_kv(0, 1, 0);
  wait_async0();
  __syncthreads();                      // Q + K slab 0 resident

  // hoist Q fragments (invariant across key slabs): 16 x v2f = 32 VGPRs
  v2f qf[16];
#pragma unroll
  for (int s = 0; s < 16; ++s)
    qf[s] = *(const v2f*)&Qst[l15 * KSTR + 4 * s + 2 * lhi];

  // ---- phase 1: S = (Q K^T) / 8 with causal + key-valid mask ----
  for (int sb = 0; sb < 16; ++sb) {
    if (sb < 15) issue_kv((sb + 1) * 128, 1, (sb + 1) & 1);

    const float* Kc = Kbuf + (size_t)(sb & 1) * 128 * KSTR;
    const int keybase = sb * 128;
    const int key     = keybase + wave * 16 + l15;
    const int kvalid  = amask[bt + key];
    v8f c = {};
#pragma unroll
    for (int s = 0; s < 16; ++s) {      // HS=64 contraction, 4 per WMMA
      v2f bb = *(const v2f*)&Kc[(wave * 16 + l15) * KSTR + 4 * s + 2 * lhi];
      c = __builtin_amdgcn_wmma_f32_16x16x4_f32(false, qf[s], false, bb,
                                                (short)0, c, false, false);
    }
#pragma unroll
    for (int r = 0; r < 8; ++r) {
      const int qrow = r + 8 * lhi;
      const bool ok  = (key <= qbase + qrow) && (kvalid != 0);
      Ssc[qrow * SSTR + key] = ok ? (c[r] * 0.125f) : -3.0e38f;
    }
    if (sb < 15) { wait_async0(); __syncthreads(); }
  }
  __syncthreads();                      // all score columns written

  // overlap the first V slab DMA with softmax + importance
  issue_kv(0, 2, 0);

  // ---- phase 2: exact row softmax (each wave owns 2 rows) ----
  for (int rr = wave * 2; rr < wave * 2 + 2; ++rr) {
    float* row = Ssc + rr * SSTR;
    float m = -3.0e38f;
    for (int j = lane; j < T_; j += 32) m = fmaxf(m, row[j]);
#pragma unroll
    for (int off = 16; off > 0; off >>= 1) m = fmaxf(m, __shfl_xor(m, off, 32));
    float sum = 0.0f;
    for (int j = lane; j < T_; j += 32) {
      float p = __expf(row[j] - m);
      row[j] = p;
      sum += p;
    }
#pragma unroll
    for (int off = 16; off > 0; off >>= 1) sum += __shfl_xor(sum, off, 32);
    const float inv = 1.0f / sum;
    for (int j = lane; j < T_; j += 32) row[j] *= inv;
  }
  __syncthreads();

  // ---- importance: sum of normalized probs over this tile's 16 queries ----
#pragma unroll
  for (int i = 0; i < 8; ++i) {
    const int key = tid + i * 256;
    float s16 = 0.0f;
#pragma unroll
    for (int r = 0; r < 16; ++r) s16 += Ssc[r * SSTR + key];
    atomicAdd(&imp[bt + key], s16 * (1.0f / ((float)H_ * (float)T_)));
  }

  // ---- phase 3: Y(16x64) = P(16x2048) @ V(2048x64) ----
  const int cb = wave & 3;              // output 16-col block (of 4)
  const int kh = wave >> 2;             // key half of each 128-key slab
  v8f cy = {};
  wait_async0();                        // V slab 0 resident
  __syncthreads();
  for (int sb = 0; sb < 16; ++sb) {
    if (sb < 15) issue_kv((sb + 1) * 128, 2, (sb + 1) & 1);

    const float* Vc = Kbuf + (size_t)(sb & 1) * 128 * KSTR;
    const int keybase = sb * 128;
#pragma unroll
    for (int s = 0; s < 16; ++s) {
      const int k4 = kh * 64 + 4 * s + 2 * lhi;        // key index within slab
      v2f a = *(const v2f*)&Ssc[l15 * SSTR + keybase + k4];
      v2f bb;
      bb.x = Vc[k4 * KSTR + cb * 16 + l15];            // B[key][d]
      bb.y = Vc[(k4 + 1) * KSTR + cb * 16 + l15];
      cy = __builtin_amdgcn_wmma_f32_16x16x4_f32(false, a, false, bb,
                                                 (short)0, cy, false, false);
    }
    if (sb < 15) { wait_async0(); __syncthreads(); }
  }

  // cross-wave reduction of the two key-halves, then store
#pragma unroll
  for (int r = 0; r < 8; ++r)
    Yred[wave * 256 + (r + 8 * lhi) * 16 + l15] = cy[r];
  __syncthreads();
#pragma unroll
  for (int i = 0; i < 4; ++i) {
    int idx = tid + i * 256;
    int row = idx >> 6, d = idx & 63;
    int c4 = d >> 4, cc = d & 15;
    float val = Yred[c4 * 256 + row * 16 + cc] + Yred[(c4 + 4) * 256 + row * 16 + cc];
    ws_y[(bt + qbase + row) * (size_t)C_ + h * HS_ + d] = val;
  }
}

// ---------------------------------------------------------------------------
// Launch: qkv GEMM -> zero importance -> attention -> proj GEMM
// Workspace: qkv (75.5 MB) + ws_y (25.2 MB)
// ---------------------------------------------------------------------------
extern "C" void kernel_launch(void* const* d_in, const int* in_sizes, int n_in,
                              void* d_out, int out_size, void* d_ws, size_t ws_size,
                              hipStream_t stream) {
  const float* x      = (const float*)d_in[0];
  const int*   amask  = (const int*)d_in[1];
  const float* w_attn = (const float*)d_in[2];
  const float* w_proj = (const float*)d_in[3];

  float* y_out = (float*)d_out;                       // (B,T,C)
  float* imp   = y_out + (size_t)B_ * T_ * C_;        // (B,T)

  float* qkv  = (float*)d_ws;                         // (B*T, 3C)
  float* ws_y = qkv + (size_t)B_ * T_ * 3 * C_;       // (B*T, C)

  const int M = B_ * T_;                              // 8192
  dim3 blk(256);

  // 1) qkv = x @ w_attn : (8192x768)*(768x2304)
  gemm_f32_wmma<<<dim3((3 * C_) / GBN, M / GBM), blk, 0, stream>>>(
      x, w_attn, qkv, M, 3 * C_, C_);

  // 2) zero importance accumulator
  zero_f32<<<dim3((B_ * T_ + 255) / 256), blk, 0, stream>>>(imp, B_ * T_);

  // 3) attention (exact softmax, LDS score slab, async double-buffered K/V)
  size_t shmem = (size_t)(16 * SSTR + 2 * 128 * KSTR + 16 * KSTR + 8 * 256)
                 * sizeof(float);                     // 213,504 B
  attn_kernel<<<dim3(T_ / 16, H_, B_), blk, shmem, stream>>>(qkv, amask, ws_y, imp);

  // 4) y_out = ws_y @ w_proj : (8192x768)*(768x768)
  gemm_f32_wmma<<<dim3(C_ / GBN, M / GBM), blk, 0, stream>>>(
      ws_y, w_proj, y_out, M, C_, C_);
}